// GNNLayer_10763188043794
// MI455X (gfx1250) — compile-verified
//
#include <hip/hip_runtime.h>

#define N_NODE   100000
#define N_EDGE   500000
#define IN_DIM   128
#define ATTN_DIM 64

typedef float v2f __attribute__((ext_vector_type(2)));
typedef float v8f __attribute__((ext_vector_type(8)));

#define WPB 2            // waves per block (500000/16 and 100000/16 both divisible by 2)
#define XS_STRIDE 132    // 128 + 4 pad: kills LDS bank conflicts, keeps 16B alignment

__device__ __forceinline__ v8f wmma_f32_4(v2f a, v2f b, v8f c) {
  // D = A(16x4 f32) x B(4x16 f32) + C(16x16 f32)
  return __builtin_amdgcn_wmma_f32_16x16x4_f32(false, a, false, b, (short)0, c, false, false);
}

// ---------------------------------------------------------------------------
// Kernel 1: per-edge  message GEMM + attention + alpha + atomic scatter to agg
// ---------------------------------------------------------------------------
__global__ __launch_bounds__(32 * WPB) void gnn_edge_kernel(
    const float* __restrict__ hidden,      // [N_NODE,128]
    const int*   __restrict__ edges,       // [N_EDGE,6]
    const float* __restrict__ h_sub,       // [N_EDGE,128]
    const float* __restrict__ rela,        // [N_REL,128]
    const float* __restrict__ W1,          // [128,128]
    const float* __restrict__ Ws_attn,     // [64,128]
    const float* __restrict__ w_alpha_w,   // [1,64]
    const float* __restrict__ w_alpha_b,   // [1]
    float* __restrict__ agg,               // [N_NODE,128] pre-zeroed (d_ws)
    float* __restrict__ alpha_out)         // [N_EDGE]
{
  __shared__ float Xs[WPB][16 * XS_STRIDE];
  __shared__ float alphaS[WPB][16];
  __shared__ int   objS[WPB][16];

  const int w     = threadIdx.x >> 5;
  const int lane  = threadIdx.x & 31;
  const int tile  = blockIdx.x * WPB + w;
  const int base  = tile * 16;             // first edge of this wave's tile
  const int n0    = lane & 15;
  const int half8 = (lane >> 4) * 8;       // row offset of lane's half in D layout
  const int koff  = (lane >> 4) * 2;       // K offset of lane's half in A/B frags

  float* Xw = Xs[w];

  if (lane < 16) objS[w][lane] = edges[(size_t)(base + lane) * 6 + 5];

  // ---- stage X = hidden[sub] + rela[rel] + h_sub  (16 edges x 128) into LDS
  for (int i = 0; i < 16; ++i) {
    const int e   = base + i;
    const int sub = edges[(size_t)e * 6 + 4];
    const int rel = edges[(size_t)e * 6 + 2];
    const float4 a = *reinterpret_cast<const float4*>(hidden + (size_t)sub * IN_DIM + lane * 4);
    const float4 b = *reinterpret_cast<const float4*>(rela   + (size_t)rel * IN_DIM + lane * 4);
    const float4 c = *reinterpret_cast<const float4*>(h_sub  + (size_t)e   * IN_DIM + lane * 4);
    float4 s;
    s.x = a.x + b.x + c.x;  s.y = a.y + b.y + c.y;
    s.z = a.z + b.z + c.z;  s.w = a.w + b.w + c.w;
    *reinterpret_cast<float4*>(&Xw[i * XS_STRIDE + lane * 4]) = s;
  }
  __syncthreads();

  // ---- GEMM1: message = X @ W1^T  -> 8 accumulator tiles of 16x16
  v8f acc[8] = {};
  for (int kk = 0; kk < 32; ++kk) {
    const int k = kk * 4 + koff;
    const v2f a = *reinterpret_cast<const v2f*>(&Xw[n0 * XS_STRIDE + k]);
    #pragma unroll
    for (int nt = 0; nt < 8; ++nt) {
      const v2f b = *reinterpret_cast<const v2f*>(W1 + (size_t)(nt * 16 + n0) * IN_DIM + k);
      acc[nt] = wmma_f32_4(a, b, acc[nt]);
    }
  }
  __syncthreads();

  // ---- spill message (D layout -> row major) into LDS for the attention GEMM
  #pragma unroll
  for (int nt = 0; nt < 8; ++nt)
    #pragma unroll
    for (int r = 0; r < 8; ++r)
      Xw[(r + half8) * XS_STRIDE + nt * 16 + n0] = acc[nt][r];
  __syncthreads();

  // ---- GEMM2: att = message @ Ws_attn^T  (16x64)
  v8f att[4] = {};
  for (int kk = 0; kk < 32; ++kk) {
    const int k = kk * 4 + koff;
    const v2f a = *reinterpret_cast<const v2f*>(&Xw[n0 * XS_STRIDE + k]);
    #pragma unroll
    for (int nt = 0; nt < 4; ++nt) {
      const v2f b = *reinterpret_cast<const v2f*>(Ws_attn + (size_t)(nt * 16 + n0) * IN_DIM + k);
      att[nt] = wmma_f32_4(a, b, att[nt]);
    }
  }

  // ---- alpha = sigmoid(relu(att) . w_alpha + bias), reduce across N (lanes)
  float wv[4];
  #pragma unroll
  for (int nt = 0; nt < 4; ++nt) wv[nt] = w_alpha_w[nt * 16 + n0];
  const float bias = w_alpha_b[0];

  #pragma unroll
  for (int r = 0; r < 8; ++r) {
    float s = 0.f;
    #pragma unroll
    for (int nt = 0; nt < 4; ++nt) s += fmaxf(att[nt][r], 0.f) * wv[nt];
    // butterfly within each 16-lane half (half = distinct row set in D layout)
    s += __shfl_xor(s, 8, 16);
    s += __shfl_xor(s, 4, 16);
    s += __shfl_xor(s, 2, 16);
    s += __shfl_xor(s, 1, 16);
    const float al = 1.f / (1.f + __expf(-(s + bias)));
    const int m = r + half8;
    if (n0 == 0) { alphaS[w][m] = al; alpha_out[base + m] = al; }
  }
  __syncthreads();

  // ---- scatter alpha * message into agg with hardware fp32 atomics
  #pragma unroll
  for (int r = 0; r < 8; ++r) {
    const int m  = r + half8;
    const float al = alphaS[w][m];
    float* dst = agg + (size_t)objS[w][m] * IN_DIM + n0;
    #pragma unroll
    for (int nt = 0; nt < 8; ++nt)
      unsafeAtomicAdd(dst + nt * 16, acc[nt][r] * al);
  }
}

// ---------------------------------------------------------------------------
// Kernel 2: hidden_new = agg @ W_h^T
// ---------------------------------------------------------------------------
__global__ __launch_bounds__(32 * WPB) void gnn_node_kernel(
    const float* __restrict__ agg,   // [N_NODE,128]
    const float* __restrict__ W_h,   // [128,128]
    float* __restrict__ out)         // [N_NODE,128]
{
  __shared__ float Xs[WPB][16 * XS_STRIDE];

  const int w     = threadIdx.x >> 5;
  const int lane  = threadIdx.x & 31;
  const int tile  = blockIdx.x * WPB + w;
  const int base  = tile * 16;             // first node of tile
  const int n0    = lane & 15;
  const int half8 = (lane >> 4) * 8;
  const int koff  = (lane >> 4) * 2;

  float* Xw = Xs[w];

  for (int i = 0; i < 16; ++i) {
    const float4 a = *reinterpret_cast<const float4*>(agg + (size_t)(base + i) * IN_DIM + lane * 4);
    *reinterpret_cast<float4*>(&Xw[i * XS_STRIDE + lane * 4]) = a;
  }
  __syncthreads();

  v8f acc[8] = {};
  for (int kk = 0; kk < 32; ++kk) {
    const int k = kk * 4 + koff;
    const v2f a = *reinterpret_cast<const v2f*>(&Xw[n0 * XS_STRIDE + k]);
    #pragma unroll
    for (int nt = 0; nt < 8; ++nt) {
      const v2f b = *reinterpret_cast<const v2f*>(W_h + (size_t)(nt * 16 + n0) * IN_DIM + k);
      acc[nt] = wmma_f32_4(a, b, acc[nt]);
    }
  }

  #pragma unroll
  for (int r = 0; r < 8; ++r) {
    float* dst = out + (size_t)(base + r + half8) * IN_DIM + n0;
    #pragma unroll
    for (int nt = 0; nt < 8; ++nt)
      dst[nt * 16] = acc[nt][r];
  }
}

extern "C" void kernel_launch(void* const* d_in, const int* in_sizes, int n_in,
                              void* d_out, int out_size, void* d_ws, size_t ws_size,
                              hipStream_t stream) {
  // setup_inputs order:
  // 0 q_sub, 1 hidden, 2 edges, 3 n_node, 4 old_nodes_new_idx, 5 entity_pretrain_emb,
  // 6 h_sub, 7 rela_embed, 8 W1, 9 Ws_attn, 10 Wr_attn(unused), 11 w_alpha_w,
  // 12 w_alpha_b, 13 W_h
  const float* hidden    = (const float*)d_in[1];
  const int*   edges     = (const int*)  d_in[2];
  const float* h_sub     = (const float*)d_in[6];
  const float* rela      = (const float*)d_in[7];
  const float* W1        = (const float*)d_in[8];
  const float* Ws_attn   = (const float*)d_in[9];
  const float* w_alpha_w = (const float*)d_in[11];
  const float* w_alpha_b = (const float*)d_in[12];
  const float* W_h       = (const float*)d_in[13];

  float* out       = (float*)d_out;                       // hidden_new [N_NODE*128]
  float* alpha_out = out + (size_t)N_NODE * IN_DIM;       // alpha [N_EDGE]
  float* agg       = (float*)d_ws;                        // [N_NODE*128] scratch

  hipMemsetAsync(agg, 0, sizeof(float) * (size_t)N_NODE * IN_DIM, stream);

  const int edge_blocks = N_EDGE / 16 / WPB;   // 15625
  const int node_blocks = N_NODE / 16 / WPB;   // 3125
  gnn_edge_kernel<<<edge_blocks, 32 * WPB, 0, stream>>>(
      hidden, edges, h_sub, rela, W1, Ws_attn, w_alpha_w, w_alpha_b, agg, alpha_out);
  gnn_node_kernel<<<node_blocks, 32 * WPB, 0, stream>>>(agg, W_h, out);
}